// OmegaSingularityModel_25984552141467
// MI455X (gfx1250) — compile-verified
//
#include <hip/hip_runtime.h>
#include <hip/hip_bf16.h>

typedef __attribute__((ext_vector_type(2))) float v2f;
typedef __attribute__((ext_vector_type(8))) float v8f;

#define CH 64

// ---------------------------------------------------------------------------
// acc[0..63] = 0
// ---------------------------------------------------------------------------
__global__ void zero_acc_kernel(float* acc) {
    if (threadIdx.x < CH) acc[threadIdx.x] = 0.0f;
}

// ---------------------------------------------------------------------------
// Pass-1 scatter: h1[dst, ch] += x[src, ch] * w[e]    (h1 pre-initialized = x)
// thread -> (edge, channel); 64 consecutive threads cover one edge's row,
// so both the gather and the atomics are fully coalesced.
// ---------------------------------------------------------------------------
__global__ void edge_scatter_kernel(const float* __restrict__ xin,
                                    const int* __restrict__ eidx,
                                    const float* __restrict__ w,
                                    float* __restrict__ h1,
                                    int nEdges) {
    long long t = (long long)blockIdx.x * blockDim.x + threadIdx.x;
    int e  = (int)(t >> 6);
    int ch = (int)(t & 63);
    if (e >= nEdges) return;
    int s = eidx[e];            // row 0 of [2,E]: src
    int d = eidx[nEdges + e];   // row 1 of [2,E]: dst
    float v = xin[(long long)s * CH + ch] * w[e];
    atomicAdd(&h1[(long long)d * CH + ch], v);
}

// ---------------------------------------------------------------------------
// Fused: h2 = relu(h1 @ W1 + b1), and acc[col] += sum_rows(h2[:, col]).
// One wave32 per 16x16 output tile, 16 x V_WMMA_F32_16X16X4_F32 over K=64.
//
// Fragment layouts (ISA 7.12.2, 32-bit, wave32):
//  A 16x4 : lane<16 -> K={0,1}, lane>=16 -> K={2,3}; M = lane&15  (float2 load)
//  B 4x16 : lane<16 -> K={0,1}, lane>=16 -> K={2,3}; N = lane&15
//  C 16x16: VGPR j -> M = j + 8*(lane>>4); N = lane&15
// ---------------------------------------------------------------------------
__global__ void __launch_bounds__(256)
gemm_relu_colsum_kernel(const float* __restrict__ h1,
                        const float* __restrict__ W1,
                        const float* __restrict__ b1,
                        float* __restrict__ h2,
                        float* __restrict__ acc,
                        int nNodes) {
    const int lane   = threadIdx.x & 31;
    const int waveId = (blockIdx.x * (blockDim.x >> 5)) + (threadIdx.x >> 5);

    const int rowTile = waveId >> 2;        // 16-row tile index
    const int colTile = waveId & 3;         // 16-col tile index (64 cols total)
    const int rowBase = rowTile * 16;
    if (rowBase >= nNodes) return;
    const int colBase = colTile * 16;

    const int mn    = lane & 15;            // M for A-frag, N for B/C-frags
    const int khalf = (lane >> 4) * 2;      // which K-pair this half-wave holds

    v8f c = {};                              // accumulator = 0

#pragma unroll
    for (int kk = 0; kk < 16; ++kk) {
        const int k0 = kk * 4;
        // A fragment: contiguous float2 from row (rowBase+mn), cols k0+khalf..+1
        v2f a = *(const v2f*)&h1[(long long)(rowBase + mn) * CH + k0 + khalf];
        // B fragment: W1[k][colBase+mn] for k = k0+khalf, k0+khalf+1
        v2f b;
        b.x = W1[(k0 + khalf)     * CH + colBase + mn];
        b.y = W1[(k0 + khalf + 1) * CH + colBase + mn];
        c = __builtin_amdgcn_wmma_f32_16x16x4_f32(
                /*neg_a=*/false, a, /*neg_b=*/false, b,
                /*c_mod=*/(short)0, c, /*reuse_a=*/false, /*reuse_b=*/false);
    }

    const int   col  = colBase + mn;
    const float bias = b1[col];
    const int   rsub = (lane >> 4) * 8;     // row offset of this half-wave
    float colSum = 0.0f;
#pragma unroll
    for (int j = 0; j < 8; ++j) {
        int row = rowBase + rsub + j;
        float v = c[j] + bias;
        v = fmaxf(v, 0.0f);                 // ReLU
        h2[(long long)row * CH + col] = v;
        colSum += v;
    }
    // fold Sigma_n h2 into acc while the tile is in registers (saves a 25.6MB pass)
    atomicAdd(&acc[col], colSum);
}

// ---------------------------------------------------------------------------
// Pass-2 collapsed: acc[ch] += Sigma_e h2[src_e, ch] * w[e]
// (scatter destination is irrelevant under the global mean -> pure reduction)
// Grid-stride; per-thread register partials -> LDS ds_add_f32 -> 64 global
// atomics per block.
// ---------------------------------------------------------------------------
__global__ void __launch_bounds__(256)
edge_reduce_kernel(const float* __restrict__ h2,
                   const int* __restrict__ eidx,
                   const float* __restrict__ w,
                   float* __restrict__ acc,
                   int nEdges) {
    __shared__ float ls[CH];
    if (threadIdx.x < CH) ls[threadIdx.x] = 0.0f;
    __syncthreads();

    const int ch    = threadIdx.x & 63;
    const int gt    = blockIdx.x * blockDim.x + threadIdx.x;
    const int egrp  = gt >> 6;                               // this thread's edge lane
    const int nGrp  = (gridDim.x * blockDim.x) >> 6;         // total edge lanes

    float local = 0.0f;
    for (int e = egrp; e < nEdges; e += nGrp) {
        int s = eidx[e];
        local += h2[(long long)s * CH + ch] * w[e];
    }
    atomicAdd(&ls[ch], local);               // ds_add_f32
    __syncthreads();
    if (threadIdx.x < CH) atomicAdd(&acc[threadIdx.x], ls[threadIdx.x]);
}

// ---------------------------------------------------------------------------
// out[c] = b2[c] + (1/N) * Sigma_k acc[k] * W2[k, c]      (64x64 matvec)
// ---------------------------------------------------------------------------
__global__ void finish_kernel(const float* __restrict__ acc,
                              const float* __restrict__ W2,
                              const float* __restrict__ b2,
                              float* __restrict__ out,
                              float invN) {
    int c = threadIdx.x;
    if (c >= CH) return;
    float s = 0.0f;
#pragma unroll
    for (int k = 0; k < CH; ++k) s = fmaf(acc[k], W2[k * CH + c], s);
    out[c] = fmaf(s, invN, b2[c]);
}

// ---------------------------------------------------------------------------
extern "C" void kernel_launch(void* const* d_in, const int* in_sizes, int n_in,
                              void* d_out, int out_size, void* d_ws, size_t ws_size,
                              hipStream_t stream) {
    const float* x    = (const float*)d_in[0];
    const int*   eidx = (const int*)  d_in[1];
    const float* w    = (const float*)d_in[2];
    const float* W1   = (const float*)d_in[3];
    const float* b1   = (const float*)d_in[4];
    const float* W2   = (const float*)d_in[5];
    const float* b2   = (const float*)d_in[6];
    float*       out  = (float*)d_out;

    const int nNodes = in_sizes[0] / CH;        // 100000
    const int nEdges = in_sizes[1] / 2;         // 1600000
    const size_t nodeElems = (size_t)nNodes * CH;

    float* h1  = (float*)d_ws;                  // [nNodes, 64]
    float* h2  = h1 + nodeElems;                // [nNodes, 64]
    float* acc = h2 + nodeElems;                // [64]

    // 1) self-loop init h1 = x ; acc = 0
    hipMemcpyAsync(h1, x, nodeElems * sizeof(float),
                   hipMemcpyDeviceToDevice, stream);
    zero_acc_kernel<<<1, 64, 0, stream>>>(acc);

    // 2) conv1 scatter: h1 += scatter_add(x[src] * w)
    {
        long long threads = (long long)nEdges * CH;
        int blocks = (int)((threads + 255) / 256);
        edge_scatter_kernel<<<blocks, 256, 0, stream>>>(x, eidx, w, h1, nEdges);
    }

    // 3) h2 = relu(h1 @ W1 + b1), acc += colsum(h2)   [WMMA f32 16x16x4]
    {
        int tiles  = ((nNodes + 15) / 16) * 4;  // 16x16 tiles over [nNodes,64]
        int blocks = (tiles + 7) / 8;           // 8 waves per 256-thread block
        gemm_relu_colsum_kernel<<<blocks, 256, 0, stream>>>(h1, W1, b1, h2, acc,
                                                            nNodes);
    }

    // 4) conv2 + mean collapsed: acc += Sigma_e h2[src]*w
    edge_reduce_kernel<<<2048, 256, 0, stream>>>(h2, eidx, w, acc, nEdges);

    // 5) out = (acc/N) @ W2 + b2
    finish_kernel<<<1, 64, 0, stream>>>(acc, W2, b2, out, 1.0f / (float)nNodes);
}